// GraphIsomorphismNetwork_44289702756677
// MI455X (gfx1250) — compile-verified
//
#include <hip/hip_runtime.h>
#include <hip/hip_bf16.h>

#define N_NODES_C   100000
#define N_EDGES_C   1600000
#define DIM_C       256
#define OUT_DIM_C   64
#define NUM_GRAPHS_C 64
#define NLAYERS_C   4
#define BN_EPS_C    1e-5f

typedef float v2f __attribute__((ext_vector_type(2)));
typedef float v8f __attribute__((ext_vector_type(8)));
typedef int v4i_t __attribute__((vector_size(4 * sizeof(int))));

// ---------------------------------------------------------------------------
// 16-byte global -> LDS copy. Uses CDNA5 async-to-LDS DMA when the toolchain
// exposes it (GLOBAL_LOAD_ASYNC_TO_LDS_B128, tracked by ASYNCcnt); otherwise
// falls back to load->store through VGPRs.
// ---------------------------------------------------------------------------
#if defined(__HIP_DEVICE_COMPILE__) && __has_builtin(__builtin_amdgcn_global_load_async_to_lds_b128)
#define GIN_ASYNC_LDS 1
#else
#define GIN_ASYNC_LDS 0
#endif

__device__ __forceinline__ void copy16_g2l(const float* __restrict__ g,
                                           float* __restrict__ l) {
#if GIN_ASYNC_LDS
    __builtin_amdgcn_global_load_async_to_lds_b128(
        (v4i_t __attribute__((address_space(1)))*)const_cast<float*>(g),
        (v4i_t __attribute__((address_space(3)))*)l,
        0, 0);
#else
    float4 v = *(const float4*)g;
    l[0] = v.x; l[1] = v.y; l[2] = v.z; l[3] = v.w;
#endif
}

__device__ __forceinline__ void copy_wait() {
#if GIN_ASYNC_LDS
#if __has_builtin(__builtin_amdgcn_s_wait_asynccnt)
    __builtin_amdgcn_s_wait_asynccnt(0);
#else
    asm volatile("s_wait_asynccnt 0x0" ::: "memory");
#endif
#endif
}

// ---------------------------------------------------------------------------
// CSR construction
// ---------------------------------------------------------------------------
__global__ __launch_bounds__(256) void hist_kernel(const int* __restrict__ src,
                                                   int* __restrict__ cnt, int nE) {
    int i = blockIdx.x * 256 + threadIdx.x;
    if (i < nE) atomicAdd(&cnt[src[i]], 1);
}

__global__ __launch_bounds__(256) void scan_block_kernel(const int* __restrict__ counts,
                                                         int* __restrict__ incl,
                                                         int* __restrict__ bsums, int n) {
    __shared__ int s[256];
    int i = blockIdx.x * 256 + threadIdx.x;
    int v = (i < n) ? counts[i] : 0;
    s[threadIdx.x] = v;
    __syncthreads();
    for (int d = 1; d < 256; d <<= 1) {
        int t = 0;
        if ((int)threadIdx.x >= d) t = s[threadIdx.x - d];
        __syncthreads();
        s[threadIdx.x] += t;
        __syncthreads();
    }
    if (i < n) incl[i] = s[threadIdx.x];
    if (threadIdx.x == 255) bsums[blockIdx.x] = s[255];
}

__global__ void scan_sums_kernel(int* __restrict__ bsums, int nb) {
    if (threadIdx.x == 0) {
        int acc = 0;
        for (int i = 0; i < nb; ++i) { int v = bsums[i]; bsums[i] = acc; acc += v; }
    }
}

__global__ __launch_bounds__(256) void scan_final_kernel(int* __restrict__ rowptr,
                                                         const int* __restrict__ counts,
                                                         const int* __restrict__ bsums,
                                                         int n, int nE) {
    int i = blockIdx.x * 256 + threadIdx.x;
    if (i < n) rowptr[i] = rowptr[i] + bsums[blockIdx.x] - counts[i];
    if (i == 0) rowptr[n] = nE;
}

__global__ __launch_bounds__(256) void fill_kernel(const int* __restrict__ src,
                                                   const int* __restrict__ dst,
                                                   int* __restrict__ cursor,
                                                   int* __restrict__ ecol, int nE) {
    int i = blockIdx.x * 256 + threadIdx.x;
    if (i < nE) {
        int p = atomicAdd(&cursor[src[i]], 1);
        ecol[p] = dst[i];
    }
}

// ---------------------------------------------------------------------------
// GIN aggregation (CSR pull): agg[i] = (1+eps)*h[i] + sum_{j in N(i)} h[j]
// block = 64 threads, one node per block, thread t handles 4 dims via float4.
// Prefetch next neighbor's row (global_prefetch_b8) to hide gather latency.
// ---------------------------------------------------------------------------
__global__ __launch_bounds__(64) void agg_kernel(const float* __restrict__ h,
                                                 const int* __restrict__ rowptr,
                                                 const int* __restrict__ ecol,
                                                 const float* __restrict__ eps,
                                                 float* __restrict__ out) {
    const int node = blockIdx.x;
    const int t4 = threadIdx.x * 4;
    const int beg = rowptr[node];
    const int end = rowptr[node + 1];
    float4 acc = make_float4(0.f, 0.f, 0.f, 0.f);
    if (beg < end) {
        int c = ecol[beg];
        for (int e = beg; e < end; ++e) {
            int cnext = (e + 1 < end) ? ecol[e + 1] : c;
            __builtin_prefetch(h + (size_t)cnext * DIM_C + t4, 0, 0);
            float4 v = *(const float4*)(h + (size_t)c * DIM_C + t4);
            acc.x += v.x; acc.y += v.y; acc.z += v.z; acc.w += v.w;
            c = cnext;
        }
    }
    const float s = 1.0f + eps[0];
    float4 hv = *(const float4*)(h + (size_t)node * DIM_C + t4);
    acc.x += s * hv.x; acc.y += s * hv.y; acc.z += s * hv.z; acc.w += s * hv.w;
    *(float4*)(out + (size_t)node * DIM_C + t4) = acc;
}

// ---------------------------------------------------------------------------
// WMMA fp32 GEMM:  Z[M x 256] = A[M x 256] @ W[256 x 256] + bias
// block = 256 threads (8 waves). Block tile: 32 (M) x 64 (N).
// Each wave owns one 16x16 C tile, loops K in steps of 4 using
// V_WMMA_F32_16X16X4_F32.  A tile (32x256) and W panel (256x64) staged in
// LDS via async-to-LDS DMA. Row strides padded to 260 / 68 floats:
// 16B-aligned for B128 bursts, bank-stride 4 -> conflict-free frag reads.
// ---------------------------------------------------------------------------
#define GM_TM 32
#define GM_TN 64
#define LDA_PAD (DIM_C + 4)   // 260 floats (1040 B row stride)
#define LDB_PAD (GM_TN + 4)   // 68 floats  (272 B row stride)

__global__ __launch_bounds__(256) void gemm_bias_kernel(const float* __restrict__ A,
                                                        const float* __restrict__ W,
                                                        const float* __restrict__ bias,
                                                        float* __restrict__ Z) {
    __shared__ __align__(16) float lA[GM_TM * LDA_PAD];
    __shared__ __align__(16) float lB[DIM_C * LDB_PAD];

    const int tid = threadIdx.x;
    const int m0 = blockIdx.x * GM_TM;
    const int n0 = blockIdx.y * GM_TN;

    // Stage A tile: 32 x 256 floats = 2048 float4, 8 per thread
    #pragma unroll
    for (int it = 0; it < 8; ++it) {
        int idx = tid + it * 256;            // 0..2047
        int row = idx >> 6;                  // /64 float4 per row
        int c4 = idx & 63;
        copy16_g2l(A + (size_t)(m0 + row) * DIM_C + c4 * 4,
                   &lA[row * LDA_PAD + c4 * 4]);
    }
    // Stage W panel: 256 x 64 floats = 4096 float4, 16 per thread
    #pragma unroll
    for (int it = 0; it < 16; ++it) {
        int idx = tid + it * 256;            // 0..4095
        int row = idx >> 4;                  // /16 float4 per row
        int c4 = idx & 15;
        copy16_g2l(W + (size_t)row * DIM_C + n0 + c4 * 4,
                   &lB[row * LDB_PAD + c4 * 4]);
    }
    copy_wait();
    __syncthreads();

    const int wave = tid >> 5;
    const int lane = tid & 31;
    const int msub = (wave >> 2) * 16;       // 0 or 16
    const int nsub = (wave & 3) * 16;        // 0,16,32,48
    const int lr = lane & 15;
    const int lh = lane >> 4;                // 0 or 1

    // C init with bias (bias depends only on N)
    v8f c;
    const float bv = bias[n0 + nsub + lr];
    #pragma unroll
    for (int r = 0; r < 8; ++r) c[r] = bv;

    const float* arow = &lA[(msub + lr) * LDA_PAD];
    const float* bcol = &lB[nsub + lr];

    #pragma unroll 8
    for (int k = 0; k < DIM_C; k += 4) {
        const int ka = k + 2 * lh;
        // A frag 16x4 f32: lane = M (lr), VGPR0/1 = K = ka, ka+1
        v2f a;
        a.x = arow[ka];
        a.y = arow[ka + 1];
        // B frag 4x16 f32: lane = N (lr), VGPR0/1 = K = ka, ka+1
        v2f b;
        b.x = bcol[(size_t)ka * LDB_PAD];
        b.y = bcol[(size_t)(ka + 1) * LDB_PAD];
        c = __builtin_amdgcn_wmma_f32_16x16x4_f32(false, a, false, b,
                                                  (short)0, c, false, false);
    }

    // Store: VGPR r, half lh -> M = msub + 8*lh + r, N = nsub + lr
    float* zp = Z + (size_t)(m0 + msub + 8 * lh) * DIM_C + n0 + nsub + lr;
    #pragma unroll
    for (int r = 0; r < 8; ++r) zp[(size_t)r * DIM_C] = c[r];
}

// ---------------------------------------------------------------------------
// Column statistics (sum, sumsq) for training-mode BatchNorm
// ---------------------------------------------------------------------------
__global__ __launch_bounds__(256) void stats_kernel(const float* __restrict__ z,
                                                    float* __restrict__ sum,
                                                    float* __restrict__ sumsq,
                                                    int n, int chunk) {
    const int t = threadIdx.x;
    const int r0 = blockIdx.x * chunk;
    const int r1 = min(n, r0 + chunk);
    float s = 0.f, q = 0.f;
    for (int r = r0; r < r1; ++r) {
        float v = z[(size_t)r * DIM_C + t];
        s += v;
        q += v * v;
    }
    atomicAdd(&sum[t], s);
    atomicAdd(&sumsq[t], q);
}

// BN (biased var) + affine + ReLU
__global__ __launch_bounds__(256) void bn_relu_kernel(const float* __restrict__ z,
                                                      const float* __restrict__ sum,
                                                      const float* __restrict__ sumsq,
                                                      const float* __restrict__ g,
                                                      const float* __restrict__ b,
                                                      float* __restrict__ out,
                                                      int n, int chunk, float inv_n) {
    const int t = threadIdx.x;
    const float mean = sum[t] * inv_n;
    const float var = sumsq[t] * inv_n - mean * mean;
    const float rstd = rsqrtf(var + BN_EPS_C);
    const float gg = g[t] * rstd;
    const float bb = b[t] - mean * gg;
    const int r0 = blockIdx.x * chunk;
    const int r1 = min(n, r0 + chunk);
    for (int r = r0; r < r1; ++r) {
        float v = z[(size_t)r * DIM_C + t] * gg + bb;
        out[(size_t)r * DIM_C + t] = fmaxf(v, 0.f);
    }
}

// ---------------------------------------------------------------------------
// Per-graph sum pooling (graph_id is sorted): flush-on-change accumulator
// ---------------------------------------------------------------------------
__global__ __launch_bounds__(256) void pool_kernel(const float* __restrict__ h,
                                                   const int* __restrict__ gid,
                                                   float* __restrict__ pooled,
                                                   int n, int chunk) {
    const int t = threadIdx.x;
    const int r0 = blockIdx.x * chunk;
    const int r1 = min(n, r0 + chunk);
    if (r0 >= r1) return;
    int cur = gid[r0];
    float acc = 0.f;
    for (int r = r0; r < r1; ++r) {
        int gg = gid[r];
        if (gg != cur) {
            atomicAdd(&pooled[(size_t)cur * DIM_C + t], acc);
            acc = 0.f;
            cur = gg;
        }
        acc += h[(size_t)r * DIM_C + t];
    }
    atomicAdd(&pooled[(size_t)cur * DIM_C + t], acc);
}

// ---------------------------------------------------------------------------
// Readout: score[g][o] = sum_l ( pooled[l][g] . lin_W[l][:,o] + lin_b[l][o] )
// ---------------------------------------------------------------------------
__global__ __launch_bounds__(64) void readout_kernel(const float* __restrict__ pooled,
                                                     const float* __restrict__ linW,
                                                     const float* __restrict__ linb,
                                                     float* __restrict__ score) {
    const int g = blockIdx.x;
    const int o = threadIdx.x;
    float acc = 0.f;
    for (int l = 0; l <= NLAYERS_C; ++l) {
        const float* p = pooled + (size_t)l * NUM_GRAPHS_C * DIM_C + (size_t)g * DIM_C;
        const float* w = linW + (size_t)l * DIM_C * OUT_DIM_C + o;
        float s = 0.f;
        #pragma unroll 4
        for (int d = 0; d < DIM_C; ++d) s += p[d] * w[(size_t)d * OUT_DIM_C];
        acc += s + linb[l * OUT_DIM_C + o];
    }
    score[(size_t)g * OUT_DIM_C + o] = acc;
}

// ---------------------------------------------------------------------------
extern "C" void kernel_launch(void* const* d_in, const int* in_sizes, int n_in,
                              void* d_out, int out_size, void* d_ws, size_t ws_size,
                              hipStream_t stream) {
    const float* x     = (const float*)d_in[0];
    const float* W1    = (const float*)d_in[1];
    const float* b1    = (const float*)d_in[2];
    const float* bn1_g = (const float*)d_in[3];
    const float* bn1_b = (const float*)d_in[4];
    const float* W2    = (const float*)d_in[5];
    const float* b2    = (const float*)d_in[6];
    const float* bn_g  = (const float*)d_in[7];
    const float* bn_b  = (const float*)d_in[8];
    const float* eps   = (const float*)d_in[9];
    const float* linW  = (const float*)d_in[10];
    const float* linb  = (const float*)d_in[11];
    const int* esrc    = (const int*)d_in[12];
    const int* edst    = (const int*)d_in[13];
    const int* gid     = (const int*)d_in[14];
    float* score       = (float*)d_out;

    // workspace carve-out
    char* ws = (char*)d_ws;
    size_t off = 0;
    auto carve = [&](size_t bytes) -> char* {
        char* p = ws + off;
        off += (bytes + 255) & ~(size_t)255;
        return p;
    };
    const size_t featBytes = sizeof(float) * (size_t)N_NODES_C * DIM_C;
    float* w0 = (float*)carve(featBytes);
    float* w1 = (float*)carve(featBytes);
    float* w2 = (float*)carve(featBytes);
    int* rowptr = (int*)carve(sizeof(int) * (N_NODES_C + 1));
    int* counts = (int*)carve(sizeof(int) * N_NODES_C);   // doubles as cursor
    int* ecol   = (int*)carve(sizeof(int) * N_EDGES_C);
    int* bsums  = (int*)carve(sizeof(int) * 1024);
    float* stats  = (float*)carve(sizeof(float) * 2 * DIM_C);
    float* pooled = (float*)carve(sizeof(float) * (NLAYERS_C + 1) * NUM_GRAPHS_C * DIM_C);
    (void)in_sizes; (void)n_in; (void)out_size; (void)ws_size;

    // ---- CSR build (edges are static; rebuilt deterministically per call) ----
    (void)hipMemsetAsync(counts, 0, sizeof(int) * N_NODES_C, stream);
    const int eblk = (N_EDGES_C + 255) / 256;
    hist_kernel<<<eblk, 256, 0, stream>>>(esrc, counts, N_EDGES_C);
    const int nblk = (N_NODES_C + 255) / 256;
    scan_block_kernel<<<nblk, 256, 0, stream>>>(counts, rowptr, bsums, N_NODES_C);
    scan_sums_kernel<<<1, 32, 0, stream>>>(bsums, nblk);
    scan_final_kernel<<<nblk, 256, 0, stream>>>(rowptr, counts, bsums, N_NODES_C, N_EDGES_C);
    (void)hipMemcpyAsync(counts, rowptr, sizeof(int) * N_NODES_C, hipMemcpyDeviceToDevice, stream);
    fill_kernel<<<eblk, 256, 0, stream>>>(esrc, edst, counts, ecol, N_EDGES_C);

    // ---- pooling of hiddens[0] = x ----
    (void)hipMemsetAsync(pooled, 0, sizeof(float) * (NLAYERS_C + 1) * NUM_GRAPHS_C * DIM_C, stream);
    const int POOL_CHUNK = 512;
    const int poolblk = (N_NODES_C + POOL_CHUNK - 1) / POOL_CHUNK;
    pool_kernel<<<poolblk, 256, 0, stream>>>(x, gid, pooled, N_NODES_C, POOL_CHUNK);

    const int STATS_CHUNK = 512;
    const int statsblk = (N_NODES_C + STATS_CHUNK - 1) / STATS_CHUNK;
    const float inv_n = 1.0f / (float)N_NODES_C;
    const dim3 ggrid(N_NODES_C / GM_TM, DIM_C / GM_TN);

    const float* hin = x;
    for (int l = 0; l < NLAYERS_C; ++l) {
        // pick two scratch buffers distinct from hin's buffer
        float *s1, *s2;
        if (hin == (const float*)w1)      { s1 = w0; s2 = w2; }
        else if (hin == (const float*)w2) { s1 = w0; s2 = w1; }
        else                              { s1 = w0; s2 = w1; }  // hin == x

        // agg = (1+eps)*h + SpMM(adj, h)        -> s1
        agg_kernel<<<N_NODES_C, 64, 0, stream>>>(hin, rowptr, ecol, eps + l, s1);
        // z1 = agg @ W1 + b1                    -> s2
        gemm_bias_kernel<<<ggrid, 256, 0, stream>>>(s1, W1 + (size_t)l * DIM_C * DIM_C,
                                                    b1 + l * DIM_C, s2);
        // a1 = relu(BN1(z1))                    -> s2 (in place)
        (void)hipMemsetAsync(stats, 0, sizeof(float) * 2 * DIM_C, stream);
        stats_kernel<<<statsblk, 256, 0, stream>>>(s2, stats, stats + DIM_C,
                                                   N_NODES_C, STATS_CHUNK);
        bn_relu_kernel<<<statsblk, 256, 0, stream>>>(s2, stats, stats + DIM_C,
                                                     bn1_g + l * DIM_C, bn1_b + l * DIM_C,
                                                     s2, N_NODES_C, STATS_CHUNK, inv_n);
        // z2 = a1 @ W2 + b2                     -> s1
        gemm_bias_kernel<<<ggrid, 256, 0, stream>>>(s2, W2 + (size_t)l * DIM_C * DIM_C,
                                                    b2 + l * DIM_C, s1);
        // h = relu(BN(z2))                      -> s2
        (void)hipMemsetAsync(stats, 0, sizeof(float) * 2 * DIM_C, stream);
        stats_kernel<<<statsblk, 256, 0, stream>>>(s1, stats, stats + DIM_C,
                                                   N_NODES_C, STATS_CHUNK);
        bn_relu_kernel<<<statsblk, 256, 0, stream>>>(s1, stats, stats + DIM_C,
                                                     bn_g + l * DIM_C, bn_b + l * DIM_C,
                                                     s2, N_NODES_C, STATS_CHUNK, inv_n);
        // pool hiddens[l+1]
        pool_kernel<<<poolblk, 256, 0, stream>>>(s2, gid,
                                                 pooled + (size_t)(l + 1) * NUM_GRAPHS_C * DIM_C,
                                                 N_NODES_C, POOL_CHUNK);
        hin = s2;
    }

    readout_kernel<<<NUM_GRAPHS_C, OUT_DIM_C, 0, stream>>>(pooled, linW, linb, score);
}